// GAT_TRANSFORMER_30760555773960
// MI455X (gfx1250) — compile-verified
//
#include <hip/hip_runtime.h>
#include <hip/hip_bf16.h>
#include <math.h>

// Problem constants (from reference)
#define N_      7
#define M_      16
#define BSN_    300000
#define MAXCOV  10.0f

// Tiling: one block handles 32 batches; 32*7 = 224 compute lanes,
// remaining 32 lanes fill the constant `scores` output.
#define BPB     32
#define THREADS 256
#define ACTIVE  (BPB * N_)   // 224
#define CAMF    (M_ * 3)     // floats of cam per batch (48)

typedef unsigned int u32;
typedef u32   __attribute__((ext_vector_type(4))) u32x4;
typedef int   __attribute__((ext_vector_type(4))) i32x4;
typedef int   __attribute__((ext_vector_type(8))) i32x8;
typedef float __attribute__((ext_vector_type(4))) f32x4;  // native vector: OK for
                                                          // __builtin_nontemporal_*

// ---------------------------------------------------------------------------
// TDM: DMA a contiguous run of `nbytes` (multiple of 4) from global memory to
// LDS using TENSOR_LOAD_TO_LDS. Descriptor packed per CDNA5 ISA §8 (D#):
//   group0: count=1, lds_addr, global_addr[56:0], type=2 ("image")
//   group1: data_size=4B, 1 x nelem tile, tensor_dim0 = stride = nelem
//   groups 2/3: zero (<=2D tensor)
// Issue once per workgroup (wave 0); completion via TENSORcnt.
// Verified lowering (round 1): tensor_load_to_lds s[0:3],s[8:15] + s_wait_tensorcnt.
// ---------------------------------------------------------------------------
__device__ __forceinline__ void tdm_load_contig_to_lds(u32 lds_byte_off,
                                                       const void* gptr,
                                                       u32 nbytes) {
  unsigned long long ga = (unsigned long long)(uintptr_t)gptr;
  u32 ne = nbytes >> 2;  // 4-byte elements

  u32x4 g0;
  g0.x = 1u;                                              // count=1, user desc
  g0.y = lds_byte_off;                                    // lds_addr
  g0.z = (u32)ga;                                         // global_addr[31:0]
  g0.w = (u32)((ga >> 32) & 0x1FFFFFFull) | 0x80000000u;  // [56:32] | type=2

  i32x8 g1;
  g1[0] = (int)(2u << 16);                         // wg_mask=0, data_size=2 (4B)
  g1[1] = (int)((ne & 0xFFFFu) << 16);             // tensor_dim0[15:0] @ 63:48
  g1[2] = (int)((ne >> 16) & 0xFFFFu) | (1 << 16); // tensor_dim0 hi, tensor_dim1=1
  g1[3] = (int)((ne & 0xFFFFu) << 16);             // tile_dim0 = ne (1536, fits 16b)
  g1[4] = 1;                                       // tile_dim1=1, tile_dim2=0
  g1[5] = (int)ne;                                 // tensor_dim0_stride[31:0]
  g1[6] = 0;
  g1[7] = 0;

  i32x4 z4 = {0, 0, 0, 0};
#if defined(__clang_major__) && (__clang_major__ >= 23)
  i32x8 z8 = {0, 0, 0, 0, 0, 0, 0, 0};
  __builtin_amdgcn_tensor_load_to_lds(g0, g1, z4, z4, z8, 0);
#else
  __builtin_amdgcn_tensor_load_to_lds(g0, g1, z4, z4, 0);
#endif
}

__global__ __launch_bounds__(THREADS) void gat_fused(
    const float* __restrict__ feat,     // (BSN*7, 8)
    const float* __restrict__ cam,      // (BSN*16, 3)
    float* __restrict__ prob_o,         // (BSN,7,16)
    float* __restrict__ pos_o,          // (BSN,7,3)
    float* __restrict__ cov_o,          // (BSN,7,1)
    float* __restrict__ scores_o,       // (BSN,8,17) = -inf
    float* __restrict__ idx_o)          // (BSN,7,1) as float
{
  __shared__ float scam[BPB * CAMF];    // 6 KB cam tile for 32 batches

  const int tid = threadIdx.x;
  const int b0  = blockIdx.x * BPB;

  // Wave 0 issues the TDM copy for the whole block; TENSORcnt is per-wave,
  // so only the issuing wave waits, then the workgroup barrier publishes LDS.
  const int wave = __builtin_amdgcn_readfirstlane((int)(tid >> 5));
  if (wave == 0) {
    typedef __attribute__((address_space(3))) float lds_f;
    u32 lds_off = (u32)(unsigned long long)(lds_f*)scam;  // workgroup-LDS offset
    tdm_load_contig_to_lds(lds_off, cam + (long long)b0 * CAMF,
                           (u32)(BPB * CAMF * sizeof(float)));
    __builtin_amdgcn_s_wait_tensorcnt(0);
  }
  __syncthreads();

  if (tid < ACTIVE) {
    const int bl = tid / N_;
    const int n  = tid - bl * N_;
    const long long row = (long long)(b0 + bl) * N_ + n;

    // feat row: 32B coalesced, read-once -> non-temporal (TH=NT) b128 loads
    const f32x4* fp = (const f32x4*)(feat + row * 8);
    const f32x4  fa = __builtin_nontemporal_load(fp);
    const f32x4  fb = __builtin_nontemporal_load(fp + 1);
    const float px = fa.x, py = fa.y, pz = fa.z;
    const float dis = fb.w;

    const float pn  = sqrtf(px*px + py*py + pz*pz);
    const float pri = 1.0f / fmaxf(pn, 1e-12f);
    const float dx = px*pri, dy = py*pri, dz = pz*pri;

    const float* cb = &scam[bl * CAMF];
    float c[M_];
    float maxv = -INFINITY;
    int   maxi = 0;
#pragma unroll
    for (int m = 0; m < M_; ++m) {
      const float cx = cb[m*3+0], cy = cb[m*3+1], cz = cb[m*3+2];
      const float n2 = cx*cx + cy*cy + cz*cz;     // ||cam|| < 1e-4  <=>  n2 < 1e-8
      float v = dx*cx + dy*cy + dz*cz;
      v = (n2 < 1e-8f) ? -INFINITY : v;           // lost-camera mask
      c[m] = v;
      if (v > maxv) { maxv = v; maxi = m; }       // first-max tiebreak like argmax
    }

    // softmax (match_cos == maxv; softmax argmax == cos argmax)
    float s = 0.0f;
#pragma unroll
    for (int m = 0; m < M_; ++m) { const float e = __expf(c[m] - maxv); c[m] = e; s += e; }
    const float is = 1.0f / s;

    // prob: 64B-aligned per row, write-once -> non-temporal b128 stores
    f32x4* po = (f32x4*)(prob_o + row * M_);
#pragma unroll
    for (int q = 0; q < 4; ++q) {
      f32x4 v4 = { c[q*4+0]*is, c[q*4+1]*is, c[q*4+2]*is, c[q*4+3]*is };
      __builtin_nontemporal_store(v4, po + q);
    }

    const bool valid = maxv > 0.99f;
    const float cov = fminf(fmaxf((1.0f - maxv) * 100.0f, 0.01f), MAXCOV);
    __builtin_nontemporal_store(valid ? cov : MAXCOV, cov_o + row);
    __builtin_nontemporal_store(valid ? (float)maxi : -1.0f, idx_o + row);

    const float mcx = cb[maxi*3+0], mcy = cb[maxi*3+1], mcz = cb[maxi*3+2];
    __builtin_nontemporal_store(valid ? dis*mcx : px, pos_o + row*3 + 0);
    __builtin_nontemporal_store(valid ? dis*mcy : py, pos_o + row*3 + 1);
    __builtin_nontemporal_store(valid ? dis*mcz : pz, pos_o + row*3 + 2);
  } else {
    // Spare 32 lanes: fill this block's scores slice (32 batches * 136 floats)
    // with -inf using coalesced non-temporal b128 stores, overlapped with
    // the compute lanes. scores is 163 MB of the 340 MB output traffic.
    const int j = tid - ACTIVE;                          // 0..31
    f32x4 v = { -INFINITY, -INFINITY, -INFINITY, -INFINITY };
    f32x4* sp = (f32x4*)(scores_o + (long long)b0 * ((N_+1)*(M_+1)));
#pragma unroll
    for (int k = 0; k < (BPB*(N_+1)*(M_+1))/(4*32); ++k)  // 34 iterations
      __builtin_nontemporal_store(v, sp + k*32 + j);
  }
}

extern "C" void kernel_launch(void* const* d_in, const int* in_sizes, int n_in,
                              void* d_out, int out_size, void* d_ws, size_t ws_size,
                              hipStream_t stream) {
  (void)in_sizes; (void)n_in; (void)out_size; (void)d_ws; (void)ws_size;

  const float* feat = (const float*)d_in[0];   // others_feat (BSN*7, 8) f32
  const float* cam  = (const float*)d_in[1];   // others_cam  (BSN*16, 3) f32

  float* out = (float*)d_out;                  // outputs concatenated flat
  const long long PROB = (long long)BSN_ * N_ * M_;          // 33,600,000
  const long long POS  = (long long)BSN_ * N_ * 3;           //  6,300,000
  const long long COV  = (long long)BSN_ * N_;               //  2,100,000
  const long long SCO  = (long long)BSN_ * (N_+1) * (M_+1);  // 40,800,000

  float* prob_o = out;
  float* pos_o  = prob_o + PROB;
  float* cov_o  = pos_o  + POS;
  float* sco_o  = cov_o  + COV;
  float* idx_o  = sco_o  + SCO;

  dim3 grid(BSN_ / BPB);   // 9375 blocks, exact cover of 300000 batches
  gat_fused<<<grid, THREADS, 0, stream>>>(feat, cam, prob_o, pos_o, cov_o,
                                          sco_o, idx_o);
}